// SurfaceField_1211180777619
// MI455X (gfx1250) — compile-verified
//
#include <hip/hip_runtime.h>
#include <hip/hip_bf16.h>

// Problem constants (from the reference file)
#define NP 32768      // points
#define NV 6890       // SMPL vertices
#define NF 13776      // faces
#define NK 9          // padded incident faces per vertex
#define VPAD 6912     // 3 * 2304, padded vertex count
#define CHUNK 2304    // vertices staged per LDS chunk (2304 * 16B = 36 KB)

typedef __attribute__((ext_vector_type(2))) float v2f;
typedef __attribute__((ext_vector_type(8))) float v8f;

// ---------------------------------------------------------------------------
// small float3 helpers (avoid relying on HIP vector operators)
// ---------------------------------------------------------------------------
struct F3 { float x, y, z; };
__device__ __forceinline__ F3 mkf3(float x, float y, float z) { F3 r; r.x = x; r.y = y; r.z = z; return r; }
__device__ __forceinline__ F3 ld3(const float* p) { return mkf3(p[0], p[1], p[2]); }
__device__ __forceinline__ F3 sub3(F3 a, F3 b) { return mkf3(a.x - b.x, a.y - b.y, a.z - b.z); }
__device__ __forceinline__ F3 scl3(F3 a, float s) { return mkf3(a.x * s, a.y * s, a.z * s); }
__device__ __forceinline__ F3 add3(F3 a, F3 b) { return mkf3(a.x + b.x, a.y + b.y, a.z + b.z); }
__device__ __forceinline__ float dot3(F3 a, F3 b) { return a.x * b.x + a.y * b.y + a.z * b.z; }
__device__ __forceinline__ F3 cross3(F3 a, F3 b) {
    return mkf3(a.y * b.z - a.z * b.y, a.z * b.x - a.x * b.z, a.x * b.y - a.y * b.x);
}
__device__ __forceinline__ float norm3(F3 a) { return sqrtf(dot3(a, a)); }
__device__ __forceinline__ float signf(float x) { return (x > 0.0f) ? 1.0f : ((x < 0.0f) ? -1.0f : 0.0f); }

// ---------------------------------------------------------------------------
// Kernel 1: face unit-normals (posed + canonical) and padded vertex table
//           vpad[i] = {x, y, z, |v|^2}, pad entries get w = 3.3e38
// ---------------------------------------------------------------------------
__global__ void sf_prep_kernel(const float* __restrict__ verts,
                               const float* __restrict__ verts0,
                               const int* __restrict__ faces,
                               float4* __restrict__ vpad,
                               float* __restrict__ fn,
                               float* __restrict__ fn0) {
    int i = blockIdx.x * blockDim.x + threadIdx.x;
    if (i < VPAD) {
        float4 o;
        if (i < NV) {
            float x = verts[3 * i + 0];
            float y = verts[3 * i + 1];
            float z = verts[3 * i + 2];
            o.x = x; o.y = y; o.z = z; o.w = x * x + y * y + z * z;
        } else {
            o.x = 0.0f; o.y = 0.0f; o.z = 0.0f; o.w = 3.3e38f;
        }
        vpad[i] = o;
    }
    if (i < NF) {
        int ia = faces[3 * i + 0];
        int ib = faces[3 * i + 1];
        int ic = faces[3 * i + 2];
        // posed mesh normal
        {
            F3 a = ld3(verts + 3 * ia), b = ld3(verts + 3 * ib), c = ld3(verts + 3 * ic);
            F3 n = cross3(sub3(b, a), sub3(c, a));
            float inv = 1.0f / norm3(n);
            fn[3 * i + 0] = n.x * inv; fn[3 * i + 1] = n.y * inv; fn[3 * i + 2] = n.z * inv;
        }
        // canonical mesh normal
        {
            F3 a = ld3(verts0 + 3 * ia), b = ld3(verts0 + 3 * ib), c = ld3(verts0 + 3 * ic);
            F3 n = cross3(sub3(b, a), sub3(c, a));
            float inv = 1.0f / norm3(n);
            fn0[3 * i + 0] = n.x * inv; fn0[3 * i + 1] = n.y * inv; fn0[3 * i + 2] = n.z * inv;
        }
    }
}

// ---------------------------------------------------------------------------
// Kernel 2: brute-force nearest vertex via V_WMMA_F32_16X16X4_F32.
// Each wave owns a 16-point tile (A matrix, constant over the V loop).
// Vertices stream through LDS in 36 KB chunks; each 16-vertex tile is one
// WMMA producing the 16x16 dot-product block. argmin over d = |v|^2 - 2*dot.
// C/D layout: slot r in lane l -> row r + 8*(l>=16), column (l&15).
// ---------------------------------------------------------------------------
__global__ void __launch_bounds__(256)
sf_nearest_kernel(const float* __restrict__ pts,
                  const float4* __restrict__ vpad,
                  int* __restrict__ nearest) {
    __shared__ float4 sv[CHUNK];

    const int lane = threadIdx.x & 31;
    const int wave = threadIdx.x >> 5;
    const int tileBase = (blockIdx.x * 8 + wave) * 16;   // 256 blocks * 8 waves * 16 pts = 32768
    const int m = lane & 15;

    // A matrix: 16x4 f32. lanes 0-15: K=0,1 (x,y); lanes 16-31: K=2,3 (z,0)
    const float px = pts[3 * (tileBase + m) + 0];
    const float py = pts[3 * (tileBase + m) + 1];
    const float pz = pts[3 * (tileBase + m) + 2];
    v2f amat;
    amat.x = (lane < 16) ? px : pz;
    amat.y = (lane < 16) ? py : 0.0f;

    float best[8];
    int   bidx[8];
#pragma unroll
    for (int r = 0; r < 8; ++r) { best[r] = 3.4e38f; bidx[r] = 0; }

    const v8f cz = {0.0f, 0.0f, 0.0f, 0.0f, 0.0f, 0.0f, 0.0f, 0.0f};

    for (int c = 0; c < VPAD / CHUNK; ++c) {
        __syncthreads();
        for (int i = threadIdx.x; i < CHUNK; i += 256)
            sv[i] = vpad[c * CHUNK + i];
        __syncthreads();

        const int cbase = c * CHUNK;
#pragma unroll 4
        for (int t = 0; t < CHUNK / 16; ++t) {
            const float4 v = sv[t * 16 + m];
            // B matrix: 4x16 f32. lanes 0-15: K=0,1 (x,y); lanes 16-31: K=2,3 (z,0)
            v2f bmat;
            bmat.x = (lane < 16) ? v.x : v.z;
            bmat.y = (lane < 16) ? v.y : 0.0f;

            v8f d = __builtin_amdgcn_wmma_f32_16x16x4_f32(
                false, amat, false, bmat, (short)0, cz, false, false);

            const int   jn  = cbase + t * 16 + m;   // vertex index for this column
            const float vsq = v.w;
#pragma unroll
            for (int r = 0; r < 8; ++r) {
                const float dist = __builtin_fmaf(-2.0f, d[r], vsq);
                const bool lt = dist < best[r];
                bidx[r] = lt ? jn : bidx[r];
                best[r] = lt ? dist : best[r];
            }
        }
    }

    // reduce (min, argmin) across the 16 lanes of each half-wave group;
    // tie-break on lowest vertex index to match jnp.argmin semantics
#pragma unroll
    for (int r = 0; r < 8; ++r) {
        float v = best[r];
        int   ix = bidx[r];
        for (int off = 8; off >= 1; off >>= 1) {
            const float ov = __shfl_xor(v, off, 16);
            const int   oi = __shfl_xor(ix, off, 16);
            const bool take = (ov < v) || (ov == v && oi < ix);
            v  = take ? ov : v;
            ix = take ? oi : ix;
        }
        best[r] = v;
        bidx[r] = ix;
    }

    if ((lane & 15) == 0) {
        const int rowOff = (lane >= 16) ? 8 : 0;
#pragma unroll
        for (int r = 0; r < 8; ++r)
            nearest[tileBase + rowOff + r] = bidx[r];
    }
}

// ---------------------------------------------------------------------------
// Kernel 3: per-point tail — face pick, plane projection, barycentric,
//           deferred projection onto the canonical mesh.
// ---------------------------------------------------------------------------
__global__ void sf_project_kernel(const float* __restrict__ pts,
                                  const float* __restrict__ verts,
                                  const float* __restrict__ verts0,
                                  const int* __restrict__ faces,
                                  const int* __restrict__ vfaces,
                                  const int* __restrict__ nearest,
                                  const float* __restrict__ fn,
                                  const float* __restrict__ fn0,
                                  float* __restrict__ out) {
    const int p = blockIdx.x * blockDim.x + threadIdx.x;
    if (p >= NP) return;

    const F3 pt = ld3(pts + 3 * p);
    const int vi = nearest[p];
    const F3 vv = ld3(verts + 3 * vi);

    const F3 look = sub3(pt, vv);
    const float look_dist = norm3(look);
    const F3 look_dir = scl3(look, 1.0f / fmaxf(look_dist, 1e-8f));

    // pick incident face maximizing |cos|  (argmin of 1-|cos|, first-min wins)
    const int* nvf = vfaces + vi * NK;
    float bestc = 3.4e38f;
    int fid = nvf[0];
    for (int k = 0; k < NK; ++k) {
        const int f = nvf[k];
        const F3 n = ld3(fn + 3 * f);
        const float cost = 1.0f - fabsf(dot3(look_dir, n));
        if (cost < bestc) { bestc = cost; fid = f; }
    }

    const F3 n = ld3(fn + 3 * fid);
    const float proj_dist = dot3(n, look);
    const F3 pp = sub3(pt, scl3(n, proj_dist));

    const int ia = faces[3 * fid + 0];
    const int ib = faces[3 * fid + 1];
    const int ic = faces[3 * fid + 2];

    const F3 t0 = ld3(verts + 3 * ia);
    const F3 t1 = ld3(verts + 3 * ib);
    const F3 t2 = ld3(verts + 3 * ic);

    // spherical barycentric weights (reference formula)
    const F3 diff0 = sub3(t0, pp), diff1 = sub3(t1, pp), diff2 = sub3(t2, pp);
    const float d0 = fmaxf(norm3(diff0), 1e-8f);
    const float d1 = fmaxf(norm3(diff1), 1e-8f);
    const float d2 = fmaxf(norm3(diff2), 1e-8f);
    const F3 u0 = scl3(diff0, 1.0f / d0);
    const F3 u1 = scl3(diff1, 1.0f / d1);
    const F3 u2 = scl3(diff2, 1.0f / d2);

    // i_left = [2,0,1], i_right = [1,2,0]; cp_i = cross(u[right], u[left])
    const F3 cp0 = cross3(u1, u2);
    const F3 cp1 = cross3(u2, u0);
    const F3 cp2 = cross3(u0, u1);

    float w0 = norm3(cp0) * d2 * d1 * signf(dot3(cp0, pp));
    float w1 = norm3(cp1) * d0 * d2 * signf(dot3(cp1, pp));
    float w2 = norm3(cp2) * d1 * d0 * signf(dot3(cp2, pp));
    const float invws = 1.0f / (w0 + w1 + w2);
    w0 *= invws; w1 *= invws; w2 *= invws;

    const F3 q0 = ld3(verts0 + 3 * ia);
    const F3 q1 = ld3(verts0 + 3 * ib);
    const F3 q2 = ld3(verts0 + 3 * ic);
    const F3 n0 = ld3(fn0 + 3 * fid);

    F3 res = add3(add3(scl3(q0, w0), scl3(q1, w1)),
                  add3(scl3(q2, w2), scl3(n0, proj_dist)));

    out[3 * p + 0] = res.x;
    out[3 * p + 1] = res.y;
    out[3 * p + 2] = res.z;
}

// ---------------------------------------------------------------------------
// Launcher
// ---------------------------------------------------------------------------
extern "C" void kernel_launch(void* const* d_in, const int* in_sizes, int n_in,
                              void* d_out, int out_size, void* d_ws, size_t ws_size,
                              hipStream_t stream) {
    const float* pts    = (const float*)d_in[0];   // [P,3]
    const float* verts  = (const float*)d_in[1];   // [V,3]
    const float* verts0 = (const float*)d_in[2];   // [V,3]
    const int*   faces  = (const int*)d_in[3];     // [F,3]
    const int*   vfaces = (const int*)d_in[4];     // [V,K]
    float*       out    = (float*)d_out;           // [P,3]

    // workspace layout (16B aligned slices)
    char* ws = (char*)d_ws;
    float4* vpad    = (float4*)(ws);                         // VPAD * 16B   = 110592
    float*  fn      = (float*)(ws + 110592);                 // NF*3*4       = 165312
    float*  fn0     = (float*)(ws + 110592 + 165312);        // NF*3*4       = 165312
    int*    nearest = (int*)(ws + 110592 + 2 * 165312);      // NP*4         = 131072

    {   // prep: covers max(VPAD, NF) = 13776 work items
        const int n = (NF > VPAD) ? NF : VPAD;
        sf_prep_kernel<<<(n + 255) / 256, 256, 0, stream>>>(verts, verts0, faces, vpad, fn, fn0);
    }
    {   // nearest: 256 blocks * 8 waves * 16 points = 32768
        sf_nearest_kernel<<<NP / 128, 256, 0, stream>>>(pts, vpad, nearest);
    }
    {   // projection tail
        sf_project_kernel<<<(NP + 255) / 256, 256, 0, stream>>>(
            pts, verts, verts0, faces, vfaces, nearest, fn, fn0, out);
    }
}